// SentinelGAT_7567732375975
// MI455X (gfx1250) — compile-verified
//
#include <hip/hip_runtime.h>

// ---------------- constants from the reference ----------------
#define F_IN   128
#define HID    64
#define HEADS  4
#define NCLS   8

typedef __attribute__((ext_vector_type(2))) float v2f;
typedef __attribute__((ext_vector_type(8))) float v8f;

// ---- monotone float<->uint encoding for atomicMax-based segment max ----
__device__ __forceinline__ unsigned encf(float x) {
  unsigned b = __float_as_uint(x);
  return (b & 0x80000000u) ? ~b : (b | 0x80000000u);
}
__device__ __forceinline__ float decf(unsigned u) {
  return (u & 0x80000000u) ? __uint_as_float(u ^ 0x80000000u)
                           : __uint_as_float(~u);
}
#define ENC_NEG_INF 0x007FFFFFu  // encf(-inf)

// ---------------- fill kernels (workspace init each call) ----------------
__global__ void gat_fill_f32(float* p, long long n, float v) {
  long long i = (long long)blockIdx.x * blockDim.x + threadIdx.x;
  if (i < n) p[i] = v;
}
__global__ void gat_fill_u32(unsigned* p, long long n, unsigned v) {
  long long i = (long long)blockIdx.x * blockDim.x + threadIdx.x;
  if (i < n) p[i] = v;
}

// ---------------- fp32 WMMA GEMM: C[M,Nout] = A[M,K] @ B[K,Nout] ----------------
// Each wave owns a 16x64 output strip: 4 accumulators share one A fragment,
// so every k-step is 1 LDS A-load + 4 B-loads + 4 v_wmma_f32_16x16x4_f32.
// The 16 x K A-tile is staged into LDS with GLOBAL_LOAD_ASYNC_TO_LDS_B32
// (ASYNCcnt-tracked, no VGPR round trip), shared by all waves of the block.
__global__ void gat_wmma_gemm_f32(const float* __restrict__ A,
                                  const float* __restrict__ B,
                                  float* __restrict__ C,
                                  int M, int K, int Nout) {
  extern __shared__ float As[];               // 16 * K floats
  const int lane   = threadIdx.x & 31;
  const int wave   = threadIdx.x >> 5;
  const int wavesB = blockDim.x >> 5;
  const int mBase  = blockIdx.x * 16;
  const int nBase0 = (blockIdx.y * wavesB + wave) * 64;

  // async stage of the A tile into LDS (row-major 16 x K), ASYNCcnt-tracked
  {
    const unsigned ldsBase = (unsigned)(uintptr_t)As;
    for (int i = threadIdx.x; i < 16 * K; i += blockDim.x) {
      int r = i / K, c = i % K;
      int row = mBase + r;
      if (row >= M) row = M - 1;              // clamp (dims divide evenly anyway)
      unsigned long long gp =
          (unsigned long long)(uintptr_t)&A[(size_t)row * K + c];
      unsigned lo = ldsBase + (unsigned)i * 4u;
      asm volatile("global_load_async_to_lds_b32 %0, %1, off"
                   :: "v"(lo), "v"(gp) : "memory");
    }
    asm volatile("s_wait_asynccnt 0x0" ::: "memory");
  }
  __syncthreads();
  if (nBase0 + 63 >= Nout + 63 - (Nout % 64) && nBase0 >= Nout) return;
  if (nBase0 >= Nout) return;

  const int m  = lane & 15;                   // M index (A) / N index (B,C,D)
  const int kh = (lane >> 4) * 2;             // K sub-offset: lanes 0-15 -> 0, 16-31 -> 2

  v8f acc0 = {}, acc1 = {}, acc2 = {}, acc3 = {};
  for (int k = 0; k < K; k += 4) {
    // A fragment: 16x4 f32, 2 VGPRs/lane (ISA 7.12.2 layout) — shared by 4 WMMAs
    v2f a;
    a.x = As[m * K + k + kh];
    a.y = As[m * K + k + kh + 1];
    const float* Brow0 = B + (size_t)(k + kh)     * Nout + m;
    const float* Brow1 = B + (size_t)(k + kh + 1) * Nout + m;
    v2f b;
    b.x = Brow0[nBase0];      b.y = Brow1[nBase0];
    acc0 = __builtin_amdgcn_wmma_f32_16x16x4_f32(false, a, false, b, (short)0, acc0, false, false);
    b.x = Brow0[nBase0 + 16]; b.y = Brow1[nBase0 + 16];
    acc1 = __builtin_amdgcn_wmma_f32_16x16x4_f32(false, a, false, b, (short)0, acc1, false, false);
    b.x = Brow0[nBase0 + 32]; b.y = Brow1[nBase0 + 32];
    acc2 = __builtin_amdgcn_wmma_f32_16x16x4_f32(false, a, false, b, (short)0, acc2, false, false);
    b.x = Brow0[nBase0 + 48]; b.y = Brow1[nBase0 + 48];
    acc3 = __builtin_amdgcn_wmma_f32_16x16x4_f32(false, a, false, b, (short)0, acc3, false, false);
  }

  // C/D layout: VGPR r -> M = r (lanes 0-15) or r+8 (lanes 16-31), N = lane%16
  const int rowOff = (lane >> 4) * 8;
#pragma unroll
  for (int r = 0; r < 8; ++r) {
    int row = mBase + rowOff + r;
    if (row < M) {
      float* Cr = C + (size_t)row * Nout + m;
      Cr[nBase0]      = acc0[r];
      Cr[nBase0 + 16] = acc1[r];
      Cr[nBase0 + 32] = acc2[r];
      Cr[nBase0 + 48] = acc3[r];
    }
  }
}

// ---------------- per-node attention dot products ----------------
__global__ void gat_att_dots(const float* __restrict__ h,
                             const float* __restrict__ att_s,
                             const float* __restrict__ att_d,
                             float* __restrict__ as_, float* __restrict__ ad_,
                             int Ntot, int H, int C) {
  int n = blockIdx.x * blockDim.x + threadIdx.x;
  if (n >= Ntot) return;
  const float* hp = h + (size_t)n * H * C;
  for (int hh = 0; hh < H; ++hh) {
    float a = 0.f, b = 0.f;
    for (int c = 0; c < C; ++c) {
      float v = hp[hh * C + c];
      a += v * att_s[hh * C + c];
      b += v * att_d[hh * C + c];
    }
    as_[(size_t)n * H + hh] = a;
    ad_[(size_t)n * H + hh] = b;
  }
}

// ---------------- edge pass 1: segment max over destinations ----------------
__global__ void gat_edge_max(const int* __restrict__ esrc, const int* __restrict__ edst,
                             int E_, int Ntot,
                             const float* __restrict__ as_, const float* __restrict__ ad_,
                             unsigned* __restrict__ m_, int H) {
  int i = blockIdx.x * blockDim.x + threadIdx.x;
  if (i >= E_ + Ntot) return;
  int s, d;
  if (i < E_) { s = esrc[i]; d = edst[i]; } else { s = d = i - E_; }
  for (int h = 0; h < H; ++h) {
    float v = as_[(size_t)s * H + h] + ad_[(size_t)d * H + h];
    v = (v > 0.f) ? v : 0.2f * v;             // LeakyReLU(0.2)
    atomicMax(&m_[(size_t)d * H + h], encf(v));
  }
}

// ---------------- edge pass 2: exp, segment sum, message scatter-add -------
__global__ void gat_edge_acc(const int* __restrict__ esrc, const int* __restrict__ edst,
                             int E_, int Ntot,
                             const float* __restrict__ as_, const float* __restrict__ ad_,
                             const unsigned* __restrict__ m_,
                             float* __restrict__ s_,
                             const float* __restrict__ hfeat,
                             float* __restrict__ agg, int H, int C) {
  int w    = blockIdx.x * (blockDim.x >> 5) + (threadIdx.x >> 5);
  int lane = threadIdx.x & 31;
  if (w >= E_ + Ntot) return;
  int s, d;
  if (w < E_) { s = esrc[w]; d = edst[w]; } else { s = d = w - E_; }

  const int HC    = H * C;
  const int per   = HC >> 5;                  // 8 (layer 1) or 2 (layer 2)
  const int chan0 = lane * per;
  const int h     = chan0 / C;                // all `per` channels share one head

  float e = as_[(size_t)s * H + h] + ad_[(size_t)d * H + h];
  e = (e > 0.f) ? e : 0.2f * e;
  float ex = expf(e - decf(m_[(size_t)d * H + h]));

  if (lane < H) {                             // one lane per head updates the sum
    float e2 = as_[(size_t)s * H + lane] + ad_[(size_t)d * H + lane];
    e2 = (e2 > 0.f) ? e2 : 0.2f * e2;
    float ex2 = expf(e2 - decf(m_[(size_t)d * H + lane]));
    atomicAdd(&s_[(size_t)d * H + lane], ex2);
  }

  const float* hs = hfeat + (size_t)s * HC;
  float*       ag = agg   + (size_t)d * HC;
#pragma unroll 8
  for (int j = 0; j < per; ++j) {
    int ch = chan0 + j;
    atomicAdd(&ag[ch], ex * hs[ch]);
  }
}

// ---------------- per-node finalize: normalize + bias + ReLU (in place) ----
__global__ void gat_finalize(float* __restrict__ agg, const float* __restrict__ s_,
                             const float* __restrict__ bias, int Ntot, int H, int C) {
  long long i   = (long long)blockIdx.x * blockDim.x + threadIdx.x;
  long long tot = (long long)Ntot * H * C;
  if (i >= tot) return;
  int HC = H * C;
  int n = (int)(i / HC), r = (int)(i % HC), h = r / C;
  float v = agg[i] / (s_[(size_t)n * H + h] + 1e-16f) + bias[r];
  agg[i] = (v > 0.f) ? v : 0.f;
}

// ---------------- global mean pool (sum + count) ----------------
__global__ void gat_pool(const float* __restrict__ hfin, const int* __restrict__ batch,
                         float* __restrict__ pooled, float* __restrict__ cnt,
                         int Ntot, int C) {
  long long i = (long long)blockIdx.x * blockDim.x + threadIdx.x;
  if (i >= (long long)Ntot * C) return;
  int n = (int)(i / C), c = (int)(i % C);
  int g = batch[n];
  atomicAdd(&pooled[(size_t)g * C + c], hfin[i]);
  if (c == 0) atomicAdd(&cnt[g], 1.0f);
}

// ---------------- classifier: (pooled/cnt) @ lin_w + lin_b ----------------
__global__ void gat_classify(const float* __restrict__ pooled, const float* __restrict__ cnt,
                             const float* __restrict__ lw, const float* __restrict__ lb,
                             float* __restrict__ out, int G, int C, int NCk) {
  int i = blockIdx.x * blockDim.x + threadIdx.x;
  if (i >= G * NCk) return;
  int g = i / NCk, k = i % NCk;
  float inv = 1.f / fmaxf(cnt[g], 1.f);
  float acc = 0.f;
  for (int c = 0; c < C; ++c) acc += pooled[(size_t)g * C + c] * lw[c * NCk + k];
  out[i] = acc * inv + lb[k];
}

// =====================================================================
extern "C" void kernel_launch(void* const* d_in, const int* in_sizes, int n_in,
                              void* d_out, int out_size, void* d_ws, size_t ws_size,
                              hipStream_t stream) {
  const float* x        = (const float*)d_in[0];
  const int*   eidx     = (const int*)  d_in[1];
  const int*   batch    = (const int*)  d_in[2];
  const float* W1       = (const float*)d_in[3];
  const float* att_src1 = (const float*)d_in[4];
  const float* att_dst1 = (const float*)d_in[5];
  const float* b1       = (const float*)d_in[6];
  const float* W2       = (const float*)d_in[7];
  const float* att_src2 = (const float*)d_in[8];
  const float* att_dst2 = (const float*)d_in[9];
  const float* b2       = (const float*)d_in[10];
  const float* lin_w    = (const float*)d_in[11];
  const float* lin_b    = (const float*)d_in[12];
  float*       out      = (float*)d_out;

  const int NN = in_sizes[0] / F_IN;     // 50000
  const int EE = in_sizes[1] / 2;        // 800000
  const int G  = out_size / NCLS;        // 512
  const int H1C = HEADS * HID;           // 256

  const int* esrc = eidx;
  const int* edst = eidx + EE;

  // ---------------- workspace carve-up (floats) ----------------
  float* ws = (float*)d_ws;
  size_t o = 0;
  float*    h1   = ws + o; o += (size_t)NN * H1C;   // x @ W1
  float*    agg1 = ws + o; o += (size_t)NN * H1C;   // layer-1 aggregate -> relu'd h
  float*    as1  = ws + o; o += (size_t)NN * HEADS;
  float*    ad1  = ws + o; o += (size_t)NN * HEADS;
  unsigned* m1   = (unsigned*)(ws + o); o += (size_t)NN * HEADS;
  float*    s1   = ws + o; o += (size_t)NN * HEADS;
  float*    h2   = ws + o; o += (size_t)NN * HID;   // relu(h1agg) @ W2
  float*    agg2 = ws + o; o += (size_t)NN * HID;
  float*    as2  = ws + o; o += (size_t)NN;
  float*    ad2  = ws + o; o += (size_t)NN;
  unsigned* m2   = (unsigned*)(ws + o); o += (size_t)NN;
  float*    s2   = ws + o; o += (size_t)NN;
  float*    pooled = ws + o; o += (size_t)G * HID;
  float*    cnt    = ws + o; o += (size_t)G;
  (void)ws_size; (void)n_in;

  const int TB = 256;
  #define GRD(n) ((int)(((long long)(n) + TB - 1) / TB))

  // ---------------- init ----------------
  gat_fill_f32<<<GRD((size_t)NN * H1C),  TB, 0, stream>>>(agg1, (long long)NN * H1C, 0.f);
  gat_fill_f32<<<GRD((size_t)NN * HEADS),TB, 0, stream>>>(s1,   (long long)NN * HEADS, 0.f);
  gat_fill_f32<<<GRD((size_t)NN * HID),  TB, 0, stream>>>(agg2, (long long)NN * HID, 0.f);
  gat_fill_f32<<<GRD(NN),                TB, 0, stream>>>(s2,   NN, 0.f);
  gat_fill_f32<<<GRD((size_t)G * HID),   TB, 0, stream>>>(pooled,(long long)G * HID, 0.f);
  gat_fill_f32<<<GRD(G),                 TB, 0, stream>>>(cnt,  G, 0.f);
  gat_fill_u32<<<GRD((size_t)NN * HEADS),TB, 0, stream>>>(m1, (long long)NN * HEADS, ENC_NEG_INF);
  gat_fill_u32<<<GRD(NN),                TB, 0, stream>>>(m2, NN, ENC_NEG_INF);

  // ---------------- layer 1 ----------------
  {   // GEMM1: h1 = x @ W1   (M=NN, K=128, N=256), fp32 WMMA, 4 waves x 16x64 strips
    dim3 grid((NN + 15) / 16, 1);
    gat_wmma_gemm_f32<<<grid, 128, 16 * F_IN * sizeof(float), stream>>>(
        x, W1, h1, NN, F_IN, H1C);
  }
  gat_att_dots<<<GRD(NN), TB, 0, stream>>>(h1, att_src1, att_dst1, as1, ad1, NN, HEADS, HID);
  gat_edge_max<<<GRD(EE + NN), TB, 0, stream>>>(esrc, edst, EE, NN, as1, ad1, m1, HEADS);
  {
    int waves = EE + NN;
    int blocks = (waves + (TB >> 5) - 1) / (TB >> 5);
    gat_edge_acc<<<blocks, TB, 0, stream>>>(esrc, edst, EE, NN, as1, ad1, m1, s1,
                                            h1, agg1, HEADS, HID);
  }
  gat_finalize<<<GRD((size_t)NN * H1C), TB, 0, stream>>>(agg1, s1, b1, NN, HEADS, HID);

  // ---------------- layer 2 ----------------
  {   // GEMM2: h2 = relu(agg1) @ W2   (M=NN, K=256, N=64), 1 wave covers 16x64
    dim3 grid((NN + 15) / 16, 1);
    gat_wmma_gemm_f32<<<grid, 32, 16 * H1C * sizeof(float), stream>>>(
        agg1, W2, h2, NN, H1C, HID);
  }
  gat_att_dots<<<GRD(NN), TB, 0, stream>>>(h2, att_src2, att_dst2, as2, ad2, NN, 1, HID);
  gat_edge_max<<<GRD(EE + NN), TB, 0, stream>>>(esrc, edst, EE, NN, as2, ad2, m2, 1);
  {
    int waves = EE + NN;
    int blocks = (waves + (TB >> 5) - 1) / (TB >> 5);
    gat_edge_acc<<<blocks, TB, 0, stream>>>(esrc, edst, EE, NN, as2, ad2, m2, s2,
                                            h2, agg2, 1, HID);
  }
  gat_finalize<<<GRD((size_t)NN * HID), TB, 0, stream>>>(agg2, s2, b2, NN, 1, HID);

  // ---------------- pool + classify ----------------
  gat_pool<<<GRD((size_t)NN * HID), TB, 0, stream>>>(agg2, batch, pooled, cnt, NN, HID);
  gat_classify<<<GRD(G * NCLS), TB, 0, stream>>>(pooled, cnt, lin_w, lin_b, out, G, HID, NCLS);
  #undef GRD
}